// Solver_20925080666149
// MI455X (gfx1250) — compile-verified
//
#include <hip/hip_runtime.h>
#include <hip/hip_bf16.h>

// CDNA5 / gfx1250, wave32. Bandwidth-bound bound-propagation objective.
// Heavy GEMVs implemented with V_WMMA_F32_16X16X4_F32; matrices streamed
// non-temporally, broadcast vectors kept cache-resident.

#define N 4096
#define TILE 16
#define NT (N / TILE)       // 256 output tiles
#define KSPLIT 16
#define KCHUNK (N / KSPLIT) // 256
#define GDIM 7

typedef __attribute__((ext_vector_type(2))) float v2f;
typedef __attribute__((ext_vector_type(8))) float v8f;

__device__ __forceinline__ v8f wmma4(v2f a, v2f b, v8f c) {
  // D = A(16x4) * B(4x16) + C(16x16), all f32
  return __builtin_amdgcn_wmma_f32_16x16x4_f32(false, a, false, b, (short)0, c,
                                               false, false);
}

// ---------------------------------------------------------------------------
// Heavy pass per iteration i: for one 16-output tile and one K-chunk compute
//   WV[j]  = sum_k W[j,k]   * Vn[k]      (A = W tile rows, B = Vn broadcast)
//   PhT[j] = sum_k Ph[k,j]  * pi[k]      (A = pi broadcast, B = Ph tile)
//   PpT[j] = sum_k Pp[k,j]  * pi[k]
// ---------------------------------------------------------------------------
__global__ __launch_bounds__(32)
void pass1_kernel(const float* __restrict__ Wi,
                  const float* __restrict__ Ph,
                  const float* __restrict__ Pp,
                  const float* __restrict__ Vn,
                  const float* __restrict__ piv,
                  float* __restrict__ partWV,
                  float* __restrict__ partPh,
                  float* __restrict__ partPp) {
  const int tile  = blockIdx.x;        // 0..NT-1
  const int split = blockIdx.y;        // 0..KSPLIT-1
  const int lane  = threadIdx.x;       // 0..31
  const int m     = lane & 15;         // row (A) / col (B) within tile
  const int koff  = (lane >> 4) << 1;  // 0 or 2: K pair owned by this half-wave
  const int base  = tile * TILE;
  const int k0    = split * KCHUNK;

  // Pointer-bump streams (constant stride per iteration).
  const float* wr  = Wi + (size_t)(base + m) * N + k0 + koff;   // +4 / iter
  const float* vn  = Vn + k0 + koff;                            // +4 / iter
  const float* pv  = piv + k0 + koff;                           // +4 / iter
  const float* phc = Ph + (size_t)(k0 + koff) * N + base + m;   // +4N / iter
  const float* ppc = Pp + (size_t)(k0 + koff) * N + base + m;   // +4N / iter

  v8f accW = {};
  v8f accA = {};
  v8f accB = {};

#pragma unroll 4
  for (int it = 0; it < KCHUNK / 4; ++it) {
    // A for W@V: two contiguous row elements per lane (b64, streamed once)
    v2f aw = __builtin_nontemporal_load((const v2f*)wr);
    // B for W@V: broadcast Vn over columns (b64, cache-resident)
    v2f bv = *(const v2f*)vn;
    accW = wmma4(aw, bv, accW);

    // A for M^T@pi: broadcast pi over rows (b64, cache-resident)
    v2f api = *(const v2f*)pv;
    // B tiles: coalesced 64B row segments, paired rows N apart (imm offset)
    v2f bph;
    bph.x = __builtin_nontemporal_load(phc);
    bph.y = __builtin_nontemporal_load(phc + N);
    accA = wmma4(api, bph, accA);

    v2f bpp;
    bpp.x = __builtin_nontemporal_load(ppc);
    bpp.y = __builtin_nontemporal_load(ppc + N);
    accB = wmma4(api, bpp, accB);

    wr += 4;
    vn += 4;
    pv += 4;
    phc += 4 * (size_t)N;
    ppc += 4 * (size_t)N;
  }

  float* oWV = partWV + ((size_t)tile * KSPLIT + split) * TILE;
  float* oPh = partPh + ((size_t)tile * KSPLIT + split) * TILE;
  float* oPp = partPp + ((size_t)tile * KSPLIT + split) * TILE;

  // accW: D[m][n] replicated over n. Lane0 holds rows 0..7 (vgpr r -> M=r),
  // lane16 holds rows 8..15 (vgpr r -> M=8+r).
  if (lane == 0) {
#pragma unroll
    for (int r = 0; r < 8; ++r) oWV[r] = accW[r];
  } else if (lane == 16) {
#pragma unroll
    for (int r = 0; r < 8; ++r) oWV[8 + r] = accW[r];
  }
  // accA/accB: D[m][n] replicated over m. Lane l<16: vgpr0 = (M=0, N=l).
  if (lane < 16) {
    oPh[lane] = accA[0];
    oPp[lane] = accB[0];
  }
}

// Final transposed GEMV: t2raw[j] = sum_k W0[k,j] * V1[k]
__global__ __launch_bounds__(32)
void pass2_kernel(const float* __restrict__ W0,
                  const float* __restrict__ V1,
                  float* __restrict__ partT2) {
  const int tile  = blockIdx.x;
  const int split = blockIdx.y;
  const int lane  = threadIdx.x;
  const int m     = lane & 15;
  const int koff  = (lane >> 4) << 1;
  const int base  = tile * TILE;
  const int k0    = split * KCHUNK;

  const float* v1  = V1 + k0 + koff;
  const float* w0c = W0 + (size_t)(k0 + koff) * N + base + m;

  v8f acc = {};
#pragma unroll 4
  for (int it = 0; it < KCHUNK / 4; ++it) {
    v2f av = *(const v2f*)v1;
    v2f bw;
    bw.x = __builtin_nontemporal_load(w0c);
    bw.y = __builtin_nontemporal_load(w0c + N);
    acc = wmma4(av, bw, acc);
    v1 += 4;
    w0c += 4 * (size_t)N;
  }
  if (lane < 16)
    partT2[((size_t)tile * KSPLIT + split) * TILE + lane] = acc[0];
}

// ---------------------------------------------------------------------------
// Light kernels
// ---------------------------------------------------------------------------
__global__ void init_kernel(float* scal) {
  if (threadIdx.x < 8) scal[threadIdx.x] = 0.0f;
}

__global__ void v4_kernel(const float* __restrict__ H,
                          const float* __restrict__ gamma,
                          float* __restrict__ V4) {
  int j = blockIdx.x * 256 + threadIdx.x;
  float s = 0.0f;
#pragma unroll
  for (int g = 0; g < GDIM; ++g) s += H[(size_t)g * N + j] * gamma[g];
  V4[j] = -s;
}

__global__ void dot_accum_kernel(const float* __restrict__ x,
                                 const float* __restrict__ y,
                                 float* accum) {
  __shared__ float red[256];
  int t = threadIdx.x;
  float s = 0.0f;
  for (int j = t; j < N; j += 256) s += x[j] * y[j];
  red[t] = s;
  __syncthreads();
  for (int off = 128; off > 0; off >>= 1) {
    if (t < off) red[t] += red[t + off];
    __syncthreads();
  }
  if (t == 0) *accum += red[0];
}

// Combine K-splits; produce vhat/WV/Pp vectors and block partials of
// S = alpha . relu(-vhat).
__global__ void reduce1_kernel(const float* __restrict__ partWV,
                               const float* __restrict__ partPh,
                               const float* __restrict__ partPp,
                               const float* __restrict__ alpha,
                               float* __restrict__ vhat,
                               float* __restrict__ WVv,
                               float* __restrict__ Ppv,
                               float* __restrict__ Spart) {
  int j = blockIdx.x * 256 + threadIdx.x;
  int tile = j >> 4, e = j & 15;
  float swv = 0.0f, sph = 0.0f, spp = 0.0f;
  for (int s = 0; s < KSPLIT; ++s) {
    size_t idx = ((size_t)tile * KSPLIT + s) * TILE + e;
    swv += partWV[idx];
    sph += partPh[idx];
    spp += partPp[idx];
  }
  float vh = swv - sph;
  vhat[j] = vh;
  WVv[j] = swv;
  Ppv[j] = spp;

  __shared__ float red[256];
  int t = threadIdx.x;
  red[t] = alpha[j] * fmaxf(-vh, 0.0f);
  __syncthreads();
  for (int off = 128; off > 0; off >>= 1) {
    if (t < off) red[t] += red[t + off];
    __syncthreads();
  }
  if (t == 0) Spart[blockIdx.x] = red[0];
}

// Per-iteration epilogue: V[i][j], dbl and sum_Vb contributions.
__global__ void finalize_kernel(const float* __restrict__ vhat,
                                const float* __restrict__ WVv,
                                const float* __restrict__ Ppv,
                                const float* __restrict__ Spart,
                                const float* __restrict__ Li,
                                const float* __restrict__ Ui,
                                const float* __restrict__ piv,
                                const float* __restrict__ pv,
                                const float* __restrict__ bi,
                                float* __restrict__ Vout,
                                float* __restrict__ scal,
                                int level) {
  __shared__ float red[256];
  __shared__ float sS, sDP;
  int t = threadIdx.x;

  red[t] = (t < 16) ? Spart[t] : 0.0f;
  __syncthreads();
  for (int off = 128; off > 0; off >>= 1) {
    if (t < off) red[t] += red[t + off];
    __syncthreads();
  }
  if (t == 0) sS = red[0];
  __syncthreads();

  float dp = 0.0f;
  for (int j = t; j < N; j += 256) dp += piv[j] * pv[j];
  red[t] = dp;
  __syncthreads();
  for (int off = 128; off > 0; off >>= 1) {
    if (t < off) red[t] += red[t + off];
    __syncthreads();
  }
  if (t == 0) sDP = red[0];
  __syncthreads();

  const float S = sS, dotpp = sDP;
  float dbl_loc = 0.0f, vb_loc = 0.0f;
  for (int j = t; j < N; j += 256) {
    float vh = vhat[j], wv = WVv[j], pp = Ppv[j];
    float Lj = Li[j], Uj = Ui[j];
    float Cj = (level == 1 && j == 1) ? 1.0f : 0.0f;
    float frac = (fmaxf(vh, 0.0f) + Uj) / (Uj - Lj);
    float unst = frac - Cj - S - pp;
    bool sa = (Lj >= 0.0f);
    bool sd = (Uj <= 0.0f);
    float v = sa ? (wv - Cj) : (sd ? -Cj : unst);
    if (!sa && !sd) dbl_loc += frac - dotpp;
    vb_loc += v * bi[j];
    Vout[j] = v;
  }

  red[t] = dbl_loc;
  __syncthreads();
  for (int off = 128; off > 0; off >>= 1) {
    if (t < off) red[t] += red[t + off];
    __syncthreads();
  }
  if (t == 0) scal[0] += red[0];  // dbl accumulator
  __syncthreads();

  red[t] = vb_loc;
  __syncthreads();
  for (int off = 128; off > 0; off >>= 1) {
    if (t < off) red[t] += red[t + off];
    __syncthreads();
  }
  if (t == 0) scal[1] += red[0];  // sum_Vb accumulator
}

// temp2[j] = -sum_k W0[k,j] V1[k]; block partials of
// relu(temp2).L0 - relu(-temp2).U0
__global__ void reduce_t2_kernel(const float* __restrict__ partT2,
                                 const float* __restrict__ L0,
                                 const float* __restrict__ U0,
                                 float* __restrict__ T2part) {
  int j = blockIdx.x * 256 + threadIdx.x;
  int tile = j >> 4, e = j & 15;
  float s = 0.0f;
  for (int sp = 0; sp < KSPLIT; ++sp)
    s += partT2[((size_t)tile * KSPLIT + sp) * TILE + e];
  float t2 = -s;
  float contrib = fmaxf(t2, 0.0f) * L0[j] - fmaxf(-t2, 0.0f) * U0[j];

  __shared__ float red[256];
  int t = threadIdx.x;
  red[t] = contrib;
  __syncthreads();
  for (int off = 128; off > 0; off >>= 1) {
    if (t < off) red[t] += red[t + off];
    __syncthreads();
  }
  if (t == 0) T2part[blockIdx.x] = red[0];
}

__global__ void grand_final_kernel(const float* __restrict__ T2part,
                                   const float* __restrict__ gamma,
                                   const float* __restrict__ d,
                                   const float* __restrict__ scal,
                                   float* __restrict__ out) {
  __shared__ float red[32];
  int t = threadIdx.x;  // blockDim == 32
  float s = (t < 16) ? T2part[t] : 0.0f;
  if (t < GDIM) s += gamma[t] * d[t];
  red[t] = s;
  __syncthreads();
  for (int off = 16; off > 0; off >>= 1) {
    if (t < off) red[t] += red[t + off];
    __syncthreads();
  }
  if (t == 0) {
    float max_obj = red[0] - scal[1] + scal[0];
    out[0] = -max_obj;
  }
}

// ---------------------------------------------------------------------------
extern "C" void kernel_launch(void* const* d_in, const int* in_sizes, int n_in,
                              void* d_out, int out_size, void* d_ws,
                              size_t ws_size, hipStream_t stream) {
  (void)in_sizes; (void)n_in; (void)out_size; (void)ws_size;
  const float* gamma = (const float*)d_in[0];
  const float* piA   = (const float*)d_in[1];   // (3, N)
  const float* alpha = (const float*)d_in[2];   // (3, N)
  const float* L     = (const float*)d_in[3];   // (4, N)
  const float* U     = (const float*)d_in[4];   // (4, N)
  const float* H     = (const float*)d_in[5];   // (7, N)
  const float* d     = (const float*)d_in[6];   // (7,)
  const float* P     = (const float*)d_in[7];   // (3, N, N)
  const float* Phat  = (const float*)d_in[8];   // (3, N, N)
  const float* p     = (const float*)d_in[9];   // (3, N)
  const float* W     = (const float*)d_in[10];  // (4, N, N)
  const float* b     = (const float*)d_in[11];  // (4, N)
  float* out = (float*)d_out;
  float* ws  = (float*)d_ws;

  // workspace layout (floats)
  float* scal   = ws;                 // [0]=dbl, [1]=sum_Vb
  float* Spart  = ws + 8;             // 16
  float* T2part = ws + 24;            // 16
  float* vhat   = ws + 64;            // N
  float* WVv    = vhat + N;           // N
  float* Ppv    = WVv + N;            // N
  float* Vbase  = Ppv + N;            // 4*N : V1..V4
  const size_t PART = (size_t)NT * KSPLIT * TILE;  // 65536
  float* partWV = Vbase + 4 * N;      // PART
  float* partPh = partWV + PART;      // PART
  float* partPp = partPh + PART;      // PART

  const dim3 hgrid(NT, KSPLIT);

  init_kernel<<<1, 32, 0, stream>>>(scal);
  v4_kernel<<<N / 256, 256, 0, stream>>>(H, gamma, Vbase + 3 * (size_t)N);
  dot_accum_kernel<<<1, 256, 0, stream>>>(Vbase + 3 * (size_t)N,
                                          b + 3 * (size_t)N, scal + 1);

  for (int i = 3; i >= 1; --i) {
    const float* Wi  = W + (size_t)i * N * N;
    const float* Phi = Phat + (size_t)(i - 1) * N * N;
    const float* Ppi = P + (size_t)(i - 1) * N * N;
    const float* piv = piA + (size_t)(i - 1) * N;
    const float* Vn  = Vbase + (size_t)i * N;        // V[i+1]
    float* Vout      = Vbase + (size_t)(i - 1) * N;  // V[i]

    pass1_kernel<<<hgrid, 32, 0, stream>>>(Wi, Phi, Ppi, Vn, piv, partWV,
                                           partPh, partPp);
    reduce1_kernel<<<N / 256, 256, 0, stream>>>(
        partWV, partPh, partPp, alpha + (size_t)(i - 1) * N, vhat, WVv, Ppv,
        Spart);
    finalize_kernel<<<1, 256, 0, stream>>>(
        vhat, WVv, Ppv, Spart, L + (size_t)i * N, U + (size_t)i * N, piv,
        p + (size_t)(i - 1) * N, b + (size_t)(i - 1) * N, Vout, scal, i);
  }

  pass2_kernel<<<hgrid, 32, 0, stream>>>(W /* W0 */, Vbase /* V1 */, partWV);
  reduce_t2_kernel<<<N / 256, 256, 0, stream>>>(partWV, L, U, T2part);
  grand_final_kernel<<<1, 32, 0, stream>>>(T2part, gamma, d, scal, out);
}